// CgpHmmCell_onedim_1314259993038
// MI455X (gfx1250) — compile-verified
//
#include <hip/hip_runtime.h>
#include <math.h>

// ---------------- problem constants (NC=2, AS=4 hardcoded as in reference) ----
#define NSTATE 24      // S
#define EMSZ   6       // EM
#define BSZ    512
#define TLEN   2048
#define BM_N   (NSTATE * EMSZ * EMSZ * EMSZ)   // 24*216 = 5184
#define BM_PAD (32 * 216)                      // padded to 32 state rows

typedef __attribute__((ext_vector_type(2))) float v2f;
typedef __attribute__((ext_vector_type(8))) float v8f;

// =======================================================================
// Setup kernel: build A [24,24] and Bm [24,6,6,6] (sparse softmaxes) in ws
//   ws[0 .. 576)          : A row-stochastic
//   ws[576 .. 576+5184)   : Bm per-state stochastic
// =======================================================================
__global__ __launch_bounds__(256) void build_tables(const float* __restrict__ tk,
                                                    const float* __restrict__ ek,
                                                    float* __restrict__ ws) {
  float* Aw = ws;
  float* Bw = ws + NSTATE * NSTATE;
  const int tid = threadIdx.x;

  // init all logits to -inf (masked positions stay -inf -> exp()==0)
  for (int i = tid; i < NSTATE * NSTATE + BM_N; i += blockDim.x) ws[i] = -INFINITY;
  __syncthreads();

  if (tid == 0) {
    // ---- scatter A logits (exact TF concatenation order, NC=2) ----
    auto SA = [&](int i, int j, float v) { Aw[i * NSTATE + j] = v; };
    SA(0, 0, 1.0f - tk[0]); SA(0, 1, tk[0]);
    SA(1, 2, 1.0f); SA(2, 3, 1.0f);
    for (int i = 0; i < 2; i++) SA(3 + 3 * i, 4 + 3 * i, tk[1 + i]);   // codon entries
    for (int i = 0; i < 2; i++) SA(4 + 3 * i, 5 + 3 * i, 1.0f);        // within-codon
    for (int i = 0; i < 2; i++) SA(5 + 3 * i, 6 + 3 * i, 1.0f);
    for (int i = 0; i < 3; i++) SA(3 + 3 * i, 14 + 3 * i, tk[3 + i]);  // inserts
    SA(9, 10, 1.0f - tk[5]);
    for (int i = 0; i < 3; i++) SA(14 + 3 * i, 15 + 3 * i, 1.0f);      // within-insert
    for (int i = 0; i < 3; i++) SA(15 + 3 * i, 16 + 3 * i, 1.0f);
    for (int i = 0; i < 3; i++) SA(16 + 3 * i, 4 + 3 * i, tk[6 + i]);  // insert->codon
    for (int i = 0; i < 3; i++) SA(16 + 3 * i, 14 + 3 * i, 1.0f - tk[6 + i]);
    float w9 = tk[9];                                                  // deletes
    SA(3, 7, 1.0f - w9 * w9);          // pw=1
    SA(3, 10, 1.0f - w9 * w9 * w9);    // pw=2
    SA(6, 10, 1.0f - w9 * w9);         // pw=1
    SA(10, 11, 1.0f); SA(11, 12, 1.0f); SA(12, 13, 1.0f);
    SA(13, 13, 1.0f); SA(13, 23, 1.0f); SA(23, 23, 1.0f);
  } else if (tid == 1) {
    // ---- scatter B logits (segments in exact reference order) ----
    auto SB = [&](int s, int i, int j, int l, float v) {
      Bw[s * 216 + i * 36 + j * 6 + l] = v;
    };
    int k = 0;
    for (int i = 0; i < 5; i++) {                       // seg: state 0
      int jm = (i + 1 > 4) ? (i + 1) : 4;
      for (int j = 0; j < jm; j++)
        for (int l = 0; l < 4; l++) SB(0, i, j, l, ek[k++]);
    }
    for (int i = 0; i < 5; i++)                         // state 1
      for (int j = 0; j < 4; j++) SB(1, i, j, 0, ek[k++]);
    for (int i = 0; i < 4; i++) SB(2, i, 0, 3, ek[k++]);
    SB(3, 0, 3, 2, 1.0f);
    for (int i = 0; i < 4; i++) SB(4, 3, 2, i, ek[k++]);
    for (int i = 0; i < 4; i++)
      for (int j = 0; j < 4; j++) SB(5, 2, i, j, ek[k++]);
    for (int s = 6; s < 10; s++)                        // codon body states
      for (int j = 0; j < 4; j++)
        for (int kk = 0; kk < 4; kk++)
          for (int l = 0; l < 4; l++) SB(s, j, kk, l, ek[k++]);
    for (int i = 0; i < 4; i++)
      for (int j = 0; j < 4; j++) SB(10, i, j, 3, ek[k++]);
    for (int i = 0; i < 4; i++) SB(11, i, 3, 0, ek[k++]);
    for (int i = 0; i < 4; i++) SB(11, i, 3, 2, ek[k++]);
    SB(12, 3, 0, 0, 1.0f); SB(12, 3, 0, 2, 1.0f); SB(12, 3, 2, 0, 1.0f);
    for (int i = 0; i < 4; i++)
      for (int j = 0; j < 4; j++)
        for (int l = 0; l < 4; l++) SB(13, i, j, l, ek[k++]);
    for (int s = 14; s < 23; s++)                       // insert states
      for (int i = 0; i < 4; i++)
        for (int j = 0; j < 4; j++)
          for (int m = 0; m < 4; m++) SB(s, i, j, m, ek[k++]);
    for (int i = 0; i < 4; i++)
      for (int j = 0; j < 4; j++) SB(23, i, j, 5, ek[k++]);
    for (int i = 0; i < 4; i++) SB(23, i, 5, 5, ek[k++]);
    SB(23, 5, 5, 5, 1.0f);
  }
  __syncthreads();

  // ---- sparse softmaxes (one row / state per thread) ----
  if (tid < NSTATE) {                       // A rows
    float mx = -INFINITY;
    for (int j = 0; j < NSTATE; j++) mx = fmaxf(mx, Aw[tid * NSTATE + j]);
    float sum = 0.0f;
    float e[NSTATE];
    for (int j = 0; j < NSTATE; j++) { e[j] = expf(Aw[tid * NSTATE + j] - mx); sum += e[j]; }
    float r = 1.0f / sum;
    for (int j = 0; j < NSTATE; j++) Aw[tid * NSTATE + j] = e[j] * r;
  } else if (tid >= 32 && tid < 32 + NSTATE) {          // B states
    int s = tid - 32;
    float mx = -INFINITY;
    for (int j = 0; j < 216; j++) mx = fmaxf(mx, Bw[s * 216 + j]);
    float sum = 0.0f;
    for (int j = 0; j < 216; j++) sum += expf(Bw[s * 216 + j] - mx);
    float r = 1.0f / sum;
    for (int j = 0; j < 216; j++) Bw[s * 216 + j] = expf(Bw[s * 216 + j] - mx) * r;
  }
}

// =======================================================================
// DPP16 butterfly sum over each 16-lane row (VALU only, no counters).
// =======================================================================
__device__ __forceinline__ float row16_sum(float x) {
  int t;
  t = __builtin_amdgcn_update_dpp(0, __float_as_int(x), 0x161, 0xF, 0xF, true); // xor 1
  x += __int_as_float(t);
  t = __builtin_amdgcn_update_dpp(0, __float_as_int(x), 0x162, 0xF, 0xF, true); // xor 2
  x += __int_as_float(t);
  t = __builtin_amdgcn_update_dpp(0, __float_as_int(x), 0x164, 0xF, 0xF, true); // xor 4
  x += __int_as_float(t);
  t = __builtin_amdgcn_update_dpp(0, __float_as_int(x), 0x168, 0xF, 0xF, true); // xor 8
  x += __int_as_float(t);
  return x;
}

// =======================================================================
// Main forward kernel: 1 wave (32 lanes) per 16-batch tile, 32 workgroups.
// Per step: f[16x24] @ A[24x24] via 12 chained V_WMMA_F32_16X16X4_F32,
// batched LDS emission gather, DPP row normalization, LDS transpose
// staging, and cooperative b128 alpha stores.
// =======================================================================
__global__ __launch_bounds__(32) void hmm_forward(const int* __restrict__ obs,
                                                  const float* __restrict__ ws,
                                                  float* __restrict__ out) {
  __shared__ float sBm[BM_PAD];      // emission table padded to 32 states (27.6 KB)
  __shared__ float sF[16 * 32];      // f staging tile (states padded to 32)
  __shared__ int   sIdx[16];         // packed (o2,o1,o0) per batch row

  const int lane = threadIdx.x;
  const int half = lane >> 4;        // which 16-lane half
  const int nlo  = lane & 15;
  const int bbase = blockIdx.x * 16;

  const float* Aw = ws;
  const float* Bw = ws + NSTATE * NSTATE;

  for (int i = lane; i < BM_N; i += 32) sBm[i] = Bw[i];
  for (int i = BM_N + lane; i < BM_PAD; i += 32) sBm[i] = 0.0f;  // pad states 24..31

  // ---- A matrix resident in WMMA B-operand layout ----
  v2f bmA0[6], bmA1[6];
  for (int c = 0; c < 6; c++) {
    for (int j = 0; j < 2; j++) {
      int row = 4 * c + j + 2 * half;
      bmA0[c][j] = Aw[row * NSTATE + nlo];
      bmA1[c][j] = (nlo < 8) ? Aw[row * NSTATE + 16 + nlo] : 0.0f;
    }
  }

  // Branchless obs context: lanes 16..31 mirror lanes 0..15 (duplicate
  // loads/stores of identical values -- removes the per-step exec branch).
  const int* rowptr = obs + (size_t)(bbase + nlo) * TLEN;
  int p1 = rowptr[0];
  int p2 = 4;                        // start context index 4
  int nxt = rowptr[1];               // pipelined fetch for t=1
  sIdx[nlo] = p1;
  __syncthreads();

  // ---- t = 0: i_vec = e0 exactly, so f0 is one-hot at state 0 ----
  float llacc[8];
  for (int v = 0; v < 8; v++) {
    int b = v + 8 * half;
    llacc[v] = __logf(sBm[4 * 36 + 4 * 6 + sIdx[b]]);      // log Bm[0,4,4,o0]
    sF[b * 32 + nlo] = (nlo == 0) ? 1.0f : 0.0f;
    sF[b * 32 + 16 + nlo] = 0.0f;
  }
  __syncthreads();
  // cooperative store of the t=0 tile (96 float4 units over 32 lanes)
  for (int j = 0; j < 3; j++) {
    int u = lane + 32 * j;
    int row = u / 6, ch = u % 6;
    float4 d = *reinterpret_cast<const float4*>(&sF[row * 32 + ch * 4]);
    float* orow = out + ((size_t)(bbase + row) * TLEN) * NSTATE;
    *reinterpret_cast<float4*>(&orow[ch * 4]) = d;
  }

  // ---- sequential scan over time ----
  for (int t = 1; t < TLEN; t++) {
    {                                 // roll the 3-symbol context (pipelined)
      int cur = nxt;
      int tn = (t + 1 < TLEN) ? (t + 1) : (TLEN - 1);
      nxt = rowptr[tn];               // latency hidden behind this whole step
      sIdx[nlo] = p2 * 36 + p1 * 6 + cur;   // lanes 16.. write identical values
      p2 = p1; p1 = cur;
    }
    // f in WMMA A-operand layout: lane -> M=nlo, V0/V1 = K=4c+2*half (+1)
    v2f fa[6];
#pragma unroll
    for (int c = 0; c < 6; c++)
      fa[c] = *reinterpret_cast<const v2f*>(&sF[nlo * 32 + 4 * c + 2 * half]);
    __syncthreads();

    // R = f @ A : chain K=4 f32 WMMAs over 6 chunks, two N-tiles
    v8f acc0 = {0, 0, 0, 0, 0, 0, 0, 0};
    v8f acc1 = {0, 0, 0, 0, 0, 0, 0, 0};
#pragma unroll
    for (int c = 0; c < 6; c++) {
      acc0 = __builtin_amdgcn_wmma_f32_16x16x4_f32(false, fa[c], false, bmA0[c],
                                                   (short)0, acc0, false, false);
      acc1 = __builtin_amdgcn_wmma_f32_16x16x4_f32(false, fa[c], false, bmA1[c],
                                                   (short)0, acc1, false, false);
    }

    // batched context indices: 8 ints per half = two ds_load_b128
    int4 ia = *reinterpret_cast<const int4*>(&sIdx[8 * half]);
    int4 ib = *reinterpret_cast<const int4*>(&sIdx[8 * half + 4]);
    int idxv[8] = {ia.x, ia.y, ia.z, ia.w, ib.x, ib.y, ib.z, ib.w};

    // issue all 16 emission gathers back-to-back (one dscnt wait total)
    float e0v[8], e1v[8];
#pragma unroll
    for (int v = 0; v < 8; v++) {
      e0v[v] = sBm[nlo * 216 + idxv[v]];
      e1v[v] = sBm[(16 + nlo) * 216 + idxv[v]];   // padded rows give 0 for s>=24
    }

    // normalize 8 independent chains (DPP butterfly + rcp/NR + native log)
#pragma unroll
    for (int v = 0; v < 8; v++) {
      float a0 = acc0[v] * e0v[v];
      float a1 = acc1[v] * e1v[v];
      float z = row16_sum(a0 + a1);
      float r = __builtin_amdgcn_rcpf(z);
      r = __fmaf_rn(r, __fmaf_rn(-z, r, 1.0f), r);   // one NR refinement
      float f0v = a0 * r, f1v = a1 * r;
      llacc[v] += __logf(z);
      int b = v + 8 * half;
      sF[b * 32 + nlo] = f0v;                  // stage for next step (24..31 stay 0)
      sF[b * 32 + 16 + nlo] = f1v;
    }
    __syncthreads();

    // cooperative, branchless alpha stores: 16 rows x 24 floats = 96 float4
    // units; 3 ds_load_b128 + 3 global_store_b128 per lane.
#pragma unroll
    for (int j = 0; j < 3; j++) {
      int u = lane + 32 * j;
      int row = u / 6, ch = u % 6;
      float4 d = *reinterpret_cast<const float4*>(&sF[row * 32 + ch * 4]);
      float* orow = out + ((size_t)(bbase + row) * TLEN + t) * NSTATE;
      *reinterpret_cast<float4*>(&orow[ch * 4]) = d;
    }
  }

  // ---- loglik (alphas occupy BSZ*TLEN*NSTATE floats first) ----
  if (nlo == 0) {
    for (int v = 0; v < 8; v++)
      out[(size_t)BSZ * TLEN * NSTATE + bbase + v + 8 * half] = llacc[v];
  }
}

// =======================================================================
extern "C" void kernel_launch(void* const* d_in, const int* in_sizes, int n_in,
                              void* d_out, int out_size, void* d_ws, size_t ws_size,
                              hipStream_t stream) {
  (void)in_sizes; (void)n_in; (void)out_size; (void)ws_size;
  const int*   obs = (const int*)d_in[0];
  const float* tk  = (const float*)d_in[1];
  const float* ek  = (const float*)d_in[2];
  // d_in[3] (init_kernel) unused: softmax of a single logit is exactly 1.
  float* ws = (float*)d_ws;

  build_tables<<<dim3(1), dim3(256), 0, stream>>>(tk, ek, ws);
  hmm_forward<<<dim3(BSZ / 16), dim3(32), 0, stream>>>(obs, ws, (float*)d_out);
}